// GNNLayer_69423851373022
// MI455X (gfx1250) — compile-verified
//
#include <hip/hip_runtime.h>

#define IN_F 256
#define OUT_F 128
#define NEG_SLOPE 0.2f

typedef __attribute__((ext_vector_type(2))) float v2f;
typedef __attribute__((ext_vector_type(8))) float v8f;

// Types for the async global->LDS DMA builtin. Clang's diagnostic revealed
// the b128 variant takes (int4 __device__*, int4 __shared__*, int, int).
typedef int v4i __attribute__((vector_size(16)));
typedef __attribute__((address_space(1))) v4i gv4i;   // global int4
typedef __attribute__((address_space(3))) v4i lv4i;   // LDS int4
typedef __attribute__((address_space(3))) void lvoid; // LDS void

// Feature tile LDS row stride in floats. 260 % 64 == 4 so the WMMA A-fragment
// reads (lanes touch {m*stride + k0 + t, t in [0,4)}) map m*4+t onto all 64
// LDS banks exactly once -> conflict-free ds_load_b64.
#define LDS_STRIDE 260
#define GEMM_GRID 2048   // persistent blocks for double-buffered streaming

// ---- gfx1250 async global->LDS copy (ASYNCcnt path) ----------------------
#if __has_builtin(__builtin_amdgcn_global_load_async_to_lds_b128)
#define ASYNC_CP_B128(gsrc, ldst)                                            \
    __builtin_amdgcn_global_load_async_to_lds_b128(                          \
        (gv4i*)(uintptr_t)(gsrc), (lv4i*)(lvoid*)(ldst), 0, 0)
#else
#define ASYNC_CP_B128(gsrc, ldst)                                            \
    asm volatile("global_load_async_to_lds_b128 %0, %1, off"                 \
                 :: "v"((unsigned)(uintptr_t)(lvoid*)(ldst)),                \
                    "v"(gsrc) : "memory")
#endif

#if __has_builtin(__builtin_amdgcn_s_wait_asynccnt)
#define WAIT_ASYNC(n) __builtin_amdgcn_s_wait_asynccnt(n)
#else
#define WAIT_ASYNC(n) asm volatile("s_wait_asynccnt " #n ::: "memory")
#endif

// -------------------------------------------------------------------------
// Kernel 1: support = LeakyReLU(features @ W^T) via V_WMMA_F32_16X16X4_F32.
// Persistent blocks, grid-stride over 16-row tiles, double-buffered LDS
// staging via async global->LDS DMA. Block = 256 threads (8 wave32); wave w
// owns the 16x16 tile at columns [16w, 16w+16). K = 256 -> 64 WMMAs/tile.
// -------------------------------------------------------------------------
__device__ __forceinline__
void async_load_tile(const float* __restrict__ feat, int rowBase, int n_nodes,
                     float* dstLds, int tid)
{
    #pragma unroll
    for (int i = 0; i < 4; ++i) {
        int idx = tid + i * 256;          // float4 slot in tile: 0..1023
        int r   = idx >> 6;               // 64 float4 per row
        int c4  = idx & 63;
        int row = rowBase + r;
        if (row >= n_nodes) row = n_nodes - 1;   // clamp (tail-safe)
        const float* src = feat + (size_t)row * IN_F + c4 * 4;
        float*       dst = dstLds + r * LDS_STRIDE + c4 * 4;
        ASYNC_CP_B128(src, dst);
    }
}

__global__ __launch_bounds__(256)
void gemm_leaky_wmma(const float* __restrict__ feat,
                     const float* __restrict__ weight,
                     float* __restrict__ support,
                     int n_nodes, int n_tiles)
{
    __shared__ float sA[2][16 * LDS_STRIDE];

    const int tid = threadIdx.x;
    int t = blockIdx.x;
    if (t >= n_tiles) return;

    const int wave    = tid >> 5;         // 0..7
    const int lane    = tid & 31;
    const int m       = lane & 15;        // A: row-in-tile / B: col-in-tile
    const int kk      = (lane >> 4) * 2;  // K sub-offset: lanes 0-15 -> 0, 16-31 -> 2
    const int colBase = wave * 16;
    const int rowOff  = (lane >> 4) << 3;

    // B[k][n] = W[n][k]; lane reads W row (colBase + m) along K (L2-resident).
    const float* wRow = weight + (size_t)(colBase + m) * IN_F;

    int buf = 0;
    async_load_tile(feat, t * 16, n_nodes, &sA[0][0], tid);

    for (; t < n_tiles; t += GEMM_GRID) {
        const int tn = t + GEMM_GRID;
        const bool hasNext = (tn < n_tiles);
        if (hasNext) {
            // prefetch next tile into the other buffer while we compute
            async_load_tile(feat, tn * 16, n_nodes, &sA[buf ^ 1][0], tid);
            WAIT_ASYNC(4);    // only the 4 just-issued DMAs may remain
        } else {
            WAIT_ASYNC(0);
        }
        __syncthreads();      // all waves' tile-t data visible in LDS

        const float* aRow = &sA[buf][m * LDS_STRIDE];
        v8f acc = {};

        #pragma unroll 8
        for (int k0 = 0; k0 < IN_F; k0 += 4) {
            float2 af = *(const float2*)(aRow + k0 + kk);   // ds_load_b64
            float2 bf = *(const float2*)(wRow + k0 + kk);   // global_load_b64
            v2f a; a.x = af.x; a.y = af.y;
            v2f b; b.x = bf.x; b.y = bf.y;
            acc = __builtin_amdgcn_wmma_f32_16x16x4_f32(
                false, a, false, b, (short)0, acc, false, false);
        }

        // D layout: VGPR r -> row = rowBase + r + 8*(lane>>4), col = colBase + m.
        const int rowBase = t * 16;
        float* sOut = support + (size_t)(rowBase + rowOff) * OUT_F + colBase + m;
        if (rowBase + 16 <= n_nodes) {            // full tile: unguarded stores
            #pragma unroll
            for (int r = 0; r < 8; ++r) {
                float x = acc[r];
                x = (x > 0.0f) ? x : NEG_SLOPE * x;
                sOut[(size_t)r * OUT_F] = x;
            }
        } else {                                  // rare tail tile
            #pragma unroll
            for (int r = 0; r < 8; ++r) {
                float x = acc[r];
                x = (x > 0.0f) ? x : NEG_SLOPE * x;
                if (rowBase + rowOff + r < n_nodes)
                    sOut[(size_t)r * OUT_F] = x;
            }
        }

        __syncthreads();      // done reading sA[buf] before it is refilled
        buf ^= 1;
    }
}

// -------------------------------------------------------------------------
// Kernel 2: zero the output accumulator.
// -------------------------------------------------------------------------
__global__ __launch_bounds__(256)
void zero_out(float4* __restrict__ out, int n4)
{
    int i = blockIdx.x * 256 + threadIdx.x;
    if (i < n4) out[i] = make_float4(0.f, 0.f, 0.f, 0.f);
}

// -------------------------------------------------------------------------
// Kernel 3: out[row] += val * support[col] for each edge.
// One wave32 per edge: lane l handles features [4l, 4l+4) -> one coalesced
// b128 gather (L2-resident) + 4 hw f32 atomics (relaxed, device scope).
// -------------------------------------------------------------------------
__global__ __launch_bounds__(256)
void spmm_scatter(const float* __restrict__ support,
                  const float* __restrict__ edge_vals,
                  const int* __restrict__ edge_rows,
                  const int* __restrict__ edge_cols,
                  float* __restrict__ out,
                  int n_edges)
{
    int wave = (blockIdx.x * 256 + threadIdx.x) >> 5;   // edge id
    int lane = threadIdx.x & 31;
    if (wave >= n_edges) return;

    int   r = edge_rows[wave];
    int   c = edge_cols[wave];
    float v = edge_vals[wave];

    float4 s = *(const float4*)(support + (size_t)c * OUT_F + lane * 4);
    float* o = out + (size_t)r * OUT_F + lane * 4;

    __hip_atomic_fetch_add(o + 0, v * s.x, __ATOMIC_RELAXED, __HIP_MEMORY_SCOPE_AGENT);
    __hip_atomic_fetch_add(o + 1, v * s.y, __ATOMIC_RELAXED, __HIP_MEMORY_SCOPE_AGENT);
    __hip_atomic_fetch_add(o + 2, v * s.z, __ATOMIC_RELAXED, __HIP_MEMORY_SCOPE_AGENT);
    __hip_atomic_fetch_add(o + 3, v * s.w, __ATOMIC_RELAXED, __HIP_MEMORY_SCOPE_AGENT);
}

// -------------------------------------------------------------------------
extern "C" void kernel_launch(void* const* d_in, const int* in_sizes, int n_in,
                              void* d_out, int out_size, void* d_ws, size_t ws_size,
                              hipStream_t stream)
{
    const float* feat   = (const float*)d_in[0];   // [N, 256]
    const float* weight = (const float*)d_in[1];   // [128, 256]
    const float* evals  = (const float*)d_in[2];   // [E]
    const int*   erows  = (const int*)d_in[3];     // [E]
    const int*   ecols  = (const int*)d_in[4];     // [E]
    float* out     = (float*)d_out;                // [N, 128]
    float* support = (float*)d_ws;                 // scratch: [N, 128] f32

    const int n_nodes = in_sizes[0] / IN_F;        // 100000
    const int n_edges = in_sizes[2];               // 3200000

    // 1) support = LeakyReLU(feat @ W^T)  -- WMMA f32, async double-buffered
    int n_tiles = (n_nodes + 15) / 16;
    int gBlocks = n_tiles < GEMM_GRID ? n_tiles : GEMM_GRID;
    gemm_leaky_wmma<<<gBlocks, 256, 0, stream>>>(feat, weight, support,
                                                 n_nodes, n_tiles);

    // 2) out = 0
    int n4 = out_size / 4;
    zero_out<<<(n4 + 255) / 256, 256, 0, stream>>>((float4*)out, n4);

    // 3) edge scatter: 8 edges per 256-thread block (1 wave32 per edge)
    int blocks = (n_edges + 7) / 8;
    spmm_scatter<<<blocks, 256, 0, stream>>>(support, evals, erows, ecols, out, n_edges);
}